// TimeFilterBlock_20633022890616
// MI455X (gfx1250) — compile-verified
//
#include <hip/hip_runtime.h>
#include <hip/hip_bf16.h>
#include <math.h>

typedef float v2f __attribute__((ext_vector_type(2)));
typedef float v8f __attribute__((ext_vector_type(8)));

#define NB    16
#define NN    1408
#define DIM   256
#define NHEAD 4
#define HDIM  64
#define KSEL  211          // max(1, int(1408*0.15))
#define NTILE (NN/16)      // 88

// ---------------------------------------------------------------------------
// Projection: z[b,h,n,:] = x[b,n,:] @ proj_w[h].T   via V_WMMA_F32_16X16X4_F32
// One wave per (i-tile, b, h); wave computes a 16x64 strip (4 accumulators).
// A fragment (16x4 f32): lane L<16 -> row M=L, K={k,k+1}; lane>=16 -> K={k+2,k+3}
// B fragment (4x16):     same addressing pattern on proj_w rows (B[kk][e]=W[e][k+kk])
// ---------------------------------------------------------------------------
__global__ __launch_bounds__(32)
void k_proj(const float* __restrict__ x, const float* __restrict__ pw,
            float* __restrict__ z) {
  const int itile = blockIdx.x;
  const int b     = blockIdx.y;
  const int h     = blockIdx.z;
  const int lane  = threadIdx.x;
  const int m     = lane & 15;
  const int koff  = (lane >> 4) * 2;
  const int mhi   = (lane >> 4) * 8;

  const float* xrow  = x  + ((size_t)b * NN + itile * 16 + m) * DIM;
  const float* wbase = pw + (size_t)h * HDIM * DIM;

  v8f acc0 = {}, acc1 = {}, acc2 = {}, acc3 = {};
  for (int k = 0; k < DIM; k += 4) {
    v2f a = *(const v2f*)(xrow + k + koff);
    v2f b0 = *(const v2f*)(wbase + (size_t)( 0 + m) * DIM + k + koff);
    v2f b1 = *(const v2f*)(wbase + (size_t)(16 + m) * DIM + k + koff);
    v2f b2 = *(const v2f*)(wbase + (size_t)(32 + m) * DIM + k + koff);
    v2f b3 = *(const v2f*)(wbase + (size_t)(48 + m) * DIM + k + koff);
    acc0 = __builtin_amdgcn_wmma_f32_16x16x4_f32(false, a, false, b0, (short)0, acc0, false, false);
    acc1 = __builtin_amdgcn_wmma_f32_16x16x4_f32(false, a, false, b1, (short)0, acc1, false, false);
    acc2 = __builtin_amdgcn_wmma_f32_16x16x4_f32(false, a, false, b2, (short)0, acc2, false, false);
    acc3 = __builtin_amdgcn_wmma_f32_16x16x4_f32(false, a, false, b3, (short)0, acc3, false, false);
  }
  float* zb = z + (((size_t)b * NHEAD + h) * NN) * HDIM;
#pragma unroll
  for (int v = 0; v < 8; ++v) {
    const int row = itile * 16 + v + mhi;
    float* zr = zb + (size_t)row * HDIM + m;
    zr[ 0] = acc0[v];
    zr[16] = acc1[v];
    zr[32] = acc2[v];
    zr[48] = acc3[v];
  }
}

// ---------------------------------------------------------------------------
// Row squared norms sq[b,h,n] = sum_d z^2
// ---------------------------------------------------------------------------
__global__ __launch_bounds__(256)
void k_sq(const float* __restrict__ z, float* __restrict__ sq) {
  const int gid = blockIdx.x * 256 + threadIdx.x;
  if (gid >= NB * NHEAD * NN) return;
  const float* zr = z + (size_t)gid * HDIM;
  float s = 0.f;
#pragma unroll 8
  for (int d = 0; d < HDIM; ++d) s += zr[d] * zr[d];
  sq[gid] = s;
}

// ---------------------------------------------------------------------------
// Distance matrix for head h: dist = sqrt(max(sq_i + sq_j - 2 z_i.z_j, 0))
// One wave per 16x16 tile of z@z^T (K=64 -> 16 WMMA f32 16x16x4 steps).
// Also emits deterministic per-block partial sums for mean(dist).
// ---------------------------------------------------------------------------
__global__ __launch_bounds__(128)
void k_dist(const float* __restrict__ z, const float* __restrict__ sq,
            float* __restrict__ dist, float* __restrict__ partial, int h) {
  const int wave = threadIdx.x >> 5;
  const int lane = threadIdx.x & 31;
  const int jt = blockIdx.x * 4 + wave;
  const int it = blockIdx.y;
  const int b  = blockIdx.z;
  const int m    = lane & 15;
  const int koff = (lane >> 4) * 2;
  const int mhi  = (lane >> 4) * 8;

  const float* zb   = z + (((size_t)b * NHEAD + h) * NN) * HDIM;
  const float* arow = zb + (size_t)(it * 16 + m) * HDIM;
  const float* brow = zb + (size_t)(jt * 16 + m) * HDIM;

  v8f acc = {};
#pragma unroll
  for (int k = 0; k < HDIM; k += 4) {
    v2f a  = *(const v2f*)(arow + k + koff);
    v2f bb = *(const v2f*)(brow + k + koff);
    acc = __builtin_amdgcn_wmma_f32_16x16x4_f32(false, a, false, bb, (short)0, acc, false, false);
  }

  const float* sqb = sq + ((size_t)b * NHEAD + h) * NN;
  const float sqj = sqb[jt * 16 + m];
  float lsum = 0.f;
#pragma unroll
  for (int v = 0; v < 8; ++v) {
    const int i = it * 16 + v + mhi;
    float d2 = sqb[i] + sqj - 2.0f * acc[v];
    float d  = (d2 > 0.f) ? sqrtf(d2) : 0.f;
    dist[((size_t)b * NN + i) * NN + jt * 16 + m] = d;
    lsum += d;
  }
  // deterministic wave reduce (fixed butterfly order)
  for (int off = 16; off > 0; off >>= 1) lsum += __shfl_xor(lsum, off, 32);
  __shared__ float wsum[4];
  if (lane == 0) wsum[wave] = lsum;
  __syncthreads();
  if (threadIdx.x == 0) {
    float s = wsum[0] + wsum[1] + wsum[2] + wsum[3];
    partial[((size_t)blockIdx.z * gridDim.y + blockIdx.y) * gridDim.x + blockIdx.x] = s;
  }
}

// Single-block deterministic reduction of the partial sums -> *S
__global__ __launch_bounds__(256)
void k_reduce(const float* __restrict__ partial, int n, float* __restrict__ S) {
  __shared__ float buf[256];
  float s = 0.f;
  for (int i = threadIdx.x; i < n; i += 256) s += partial[i];
  buf[threadIdx.x] = s;
  __syncthreads();
  for (int off = 128; off > 0; off >>= 1) {
    if (threadIdx.x < off) buf[threadIdx.x] += buf[threadIdx.x + off];
    __syncthreads();
  }
  if (threadIdx.x == 0) *S = buf[0];
}

// ---------------------------------------------------------------------------
// Per-row k-th smallest distance (threshold t_i) via 31-step binary search
// over float bit patterns (all distances >= 0 so uint order == float order).
// One block per (row, batch); row staged in LDS.
// ---------------------------------------------------------------------------
__global__ __launch_bounds__(256)
void k_kth(const float* __restrict__ dist, float* __restrict__ tval) {
  __shared__ unsigned sbits[NN];
  __shared__ int cnt[256];
  const int row = blockIdx.x;
  const int b   = blockIdx.y;
  const float* dr = dist + ((size_t)b * NN + row) * NN;
  for (int j = threadIdx.x; j < NN; j += 256) sbits[j] = __float_as_uint(dr[j]);
  __syncthreads();
  unsigned lo = 0u, hi = 0x7F800000u;
  while (lo < hi) {
    const unsigned mid = lo + ((hi - lo) >> 1);
    int c = 0;
    for (int j = threadIdx.x; j < NN; j += 256) c += (sbits[j] <= mid) ? 1 : 0;
    cnt[threadIdx.x] = c;
    __syncthreads();
    for (int off = 128; off > 0; off >>= 1) {
      if (threadIdx.x < off) cnt[threadIdx.x] += cnt[threadIdx.x + off];
      __syncthreads();
    }
    const int total = cnt[0];
    __syncthreads();
    if (total >= KSEL) hi = mid; else lo = mid + 1;
  }
  if (threadIdx.x == 0) tval[(size_t)b * NN + row] = __uint_as_float(lo);
}

// ---------------------------------------------------------------------------
// adj accumulate: adj += exp(-d^2/(2*mean^2+1e-8)) * (d <= max(t_i,t_j))
// (symmetric-OR of the kNN mask folds into max of row thresholds)
// ---------------------------------------------------------------------------
__global__ __launch_bounds__(256)
void k_adj(const float* __restrict__ dist, const float* __restrict__ tval,
           const float* __restrict__ S, float* __restrict__ adj, int first) {
  const size_t gid = (size_t)blockIdx.x * 256 + threadIdx.x;
  const size_t total = (size_t)NB * NN * (NN / 4);
  if (gid >= total) return;
  const int    j4 = (int)(gid % (NN / 4));
  const size_t bi = gid / (NN / 4);
  const int    i  = (int)(bi % NN);
  const int    b  = (int)(bi / NN);

  const float mean = (*S) * (1.0f / ((float)NB * (float)NN * (float)NN));
  const float inv  = 1.0f / (2.0f * mean * mean + 1e-8f);

  const float4 d4 = *(const float4*)(dist + bi * NN + (size_t)j4 * 4);
  const float ti = tval[(size_t)b * NN + i];
  const float* tj = tval + (size_t)b * NN + (size_t)j4 * 4;
  float dv[4] = {d4.x, d4.y, d4.z, d4.w};
  float o[4];
#pragma unroll
  for (int q = 0; q < 4; ++q) {
    const float d = dv[q];
    const float sim = expf(-d * d * inv);
    o[q] = (d <= fmaxf(ti, tj[q])) ? sim : 0.f;
  }
  float4* ap = (float4*)(adj + bi * NN + (size_t)j4 * 4);
  if (first) {
    *ap = make_float4(o[0], o[1], o[2], o[3]);
  } else {
    float4 p = *ap;
    *ap = make_float4(p.x + o[0], p.y + o[1], p.z + o[2], p.w + o[3]);
  }
}

// ---------------------------------------------------------------------------
// Per-node: score MLP (256->128 gelu ->1 sigmoid) and gate softmax + top-p.
// One 128-thread block per node; x row staged in LDS.
// ---------------------------------------------------------------------------
__global__ __launch_bounds__(128)
void k_node(const float* __restrict__ x,
            const float* __restrict__ w1, const float* __restrict__ b1,
            const float* __restrict__ w2, const float* __restrict__ b2,
            const float* __restrict__ gw, const float* __restrict__ gb,
            float* __restrict__ scores, float* __restrict__ wvec) {
  __shared__ float xs[DIM];
  __shared__ float red[128];
  __shared__ float logits[3];
  const int node = blockIdx.x;
  const int t = threadIdx.x;
  const float* xr = x + (size_t)node * DIM;
  xs[t]       = xr[t];
  xs[t + 128] = xr[t + 128];
  __syncthreads();

  float acc = 0.f;
  const float* wr = w1 + (size_t)t * DIM;
  for (int d = 0; d < DIM; ++d) acc += xs[d] * wr[d];
  acc += b1[t];
  const float g = 0.5f * acc * (1.0f + erff(acc * 0.70710678118654752f)); // exact gelu
  red[t] = g * w2[t];
  if (t < 3) {
    const float* gr = gw + (size_t)t * DIM;
    float a = 0.f;
    for (int d = 0; d < DIM; ++d) a += xs[d] * gr[d];
    logits[t] = a + gb[t];
  }
  __syncthreads();
  for (int off = 64; off > 0; off >>= 1) {
    if (t < off) red[t] += red[t + off];
    __syncthreads();
  }
  if (t == 0) {
    scores[node] = 1.0f / (1.0f + expf(-(red[0] + b2[0])));
    // softmax over 3 gate logits
    const float mx = fmaxf(logits[0], fmaxf(logits[1], logits[2]));
    const float e0 = expf(logits[0] - mx), e1 = expf(logits[1] - mx), e2 = expf(logits[2] - mx);
    const float Z = e0 + e1 + e2;
    float p[3] = {e0 / Z, e1 / Z, e2 / Z};
    // stable descending order of 3 (strict >, ties keep index order)
    int o0 = 0, o1 = 1, o2 = 2, tmp;
    if (p[o1] > p[o0]) { tmp = o0; o0 = o1; o1 = tmp; }
    if (p[o2] > p[o1]) { tmp = o1; o1 = o2; o2 = tmp; }
    if (p[o1] > p[o0]) { tmp = o0; o0 = o1; o1 = tmp; }
    const float sp0 = p[o0], sp1 = p[o1], sp2 = p[o2];
    float w[3];
    w[o0] = sp0;                                     // cum-before = 0 < 0.85
    w[o1] = (sp0 < 0.85f) ? sp1 : 0.f;
    w[o2] = (sp0 + sp1 < 0.85f) ? sp2 : 0.f;
    const float s = w[0] + w[1] + w[2] + 1e-8f;
    wvec[(size_t)node * 3 + 0] = w[0] / s;
    wvec[(size_t)node * 3 + 1] = w[1] / s;
    wvec[(size_t)node * 3 + 2] = w[2] / s;
  }
}

// ---------------------------------------------------------------------------
// Final fused output:
// out = (adj/H) * [ tm*avg(w0) + sm*avg(w1) + sigmoid((s_i s_j - th)*10)*avg(w2) ]
// ---------------------------------------------------------------------------
__global__ __launch_bounds__(256)
void k_final(const float* __restrict__ adj, const float* __restrict__ tm,
             const float* __restrict__ sm, const float* __restrict__ scores,
             const float* __restrict__ wvec, const float* __restrict__ thr,
             float* __restrict__ out) {
  const size_t gid = (size_t)blockIdx.x * 256 + threadIdx.x;
  const size_t total = (size_t)NB * NN * (NN / 4);
  if (gid >= total) return;
  const int    j4 = (int)(gid % (NN / 4));
  const size_t bi = gid / (NN / 4);
  const int    i  = (int)(bi % NN);
  const int    b  = (int)(bi / NN);

  const float4 a4 = *(const float4*)(adj + bi * NN + (size_t)j4 * 4);
  const float4 t4 = *(const float4*)(tm + (size_t)i * NN + (size_t)j4 * 4);
  const float4 s4 = *(const float4*)(sm + (size_t)i * NN + (size_t)j4 * 4);
  const float si  = scores[(size_t)b * NN + i];
  const float* wi = wvec + ((size_t)b * NN + i) * 3;
  const float th  = thr[0];

  float av[4] = {a4.x, a4.y, a4.z, a4.w};
  float tv[4] = {t4.x, t4.y, t4.z, t4.w};
  float sv[4] = {s4.x, s4.y, s4.z, s4.w};
  float ov[4];
#pragma unroll
  for (int q = 0; q < 4; ++q) {
    const int j = j4 * 4 + q;
    const float sj = scores[(size_t)b * NN + j];
    const float* wj = wvec + ((size_t)b * NN + j) * 3;
    const float pgate = 1.0f / (1.0f + expf(-(si * sj - th) * 10.0f));
    const float mix = tv[q] * 0.5f * (wi[0] + wj[0])
                    + sv[q] * 0.5f * (wi[1] + wj[1])
                    + pgate * 0.5f * (wi[2] + wj[2]);
    ov[q] = av[q] * 0.25f * mix;   // /H folded here
  }
  *(float4*)(out + bi * NN + (size_t)j4 * 4) = make_float4(ov[0], ov[1], ov[2], ov[3]);
}

// ---------------------------------------------------------------------------
extern "C" void kernel_launch(void* const* d_in, const int* in_sizes, int n_in,
                              void* d_out, int out_size, void* d_ws, size_t ws_size,
                              hipStream_t stream) {
  const float* x        = (const float*)d_in[0];
  const float* proj_w   = (const float*)d_in[1];
  const float* gate_w   = (const float*)d_in[2];
  const float* gate_b   = (const float*)d_in[3];
  const float* score_w1 = (const float*)d_in[4];
  const float* score_b1 = (const float*)d_in[5];
  const float* score_w2 = (const float*)d_in[6];
  const float* score_b2 = (const float*)d_in[7];
  const float* thr      = (const float*)d_in[8];
  const float* tmask    = (const float*)d_in[9];
  const float* smask    = (const float*)d_in[10];
  float* out = (float*)d_out;

  float* w = (float*)d_ws;
  size_t off = 0;
  float* z      = w + off; off += (size_t)NB * NHEAD * NN * HDIM; // 5,767,168
  float* sq     = w + off; off += (size_t)NB * NHEAD * NN;        //    90,112
  float* tval   = w + off; off += (size_t)NB * NN;                //    22,528
  float* scores = w + off; off += (size_t)NB * NN;                //    22,528
  float* wvec   = w + off; off += (size_t)NB * NN * 3;            //    67,584
  float* part   = w + off; off += 32768;
  float* S      = w + off; off += 8;
  float* dist   = w + off; off += (size_t)NB * NN * NN;           // 31,719,424
  float* adj    = w + off; off += (size_t)NB * NN * NN;           // 31,719,424

  // Projection + norms + per-node score/gate (independent of heads)
  k_proj<<<dim3(NTILE, NB, NHEAD), 32, 0, stream>>>(x, proj_w, z);
  {
    const int n = NB * NHEAD * NN;
    k_sq<<<(n + 255) / 256, 256, 0, stream>>>(z, sq);
  }
  k_node<<<NB * NN, 128, 0, stream>>>(x, score_w1, score_b1, score_w2, score_b2,
                                      gate_w, gate_b, scores, wvec);

  const int nPart = (NTILE / 4) * NTILE * NB;           // 22*88*16 = 30976
  const int nElemBlocks = (int)(((size_t)NB * NN * (NN / 4) + 255) / 256); // 30976

  for (int h = 0; h < NHEAD; ++h) {
    k_dist<<<dim3(NTILE / 4, NTILE, NB), 128, 0, stream>>>(z, sq, dist, part, h);
    k_reduce<<<1, 256, 0, stream>>>(part, nPart, S);
    k_kth<<<dim3(NN, NB), 256, 0, stream>>>(dist, tval);
    k_adj<<<nElemBlocks, 256, 0, stream>>>(dist, tval, S, adj, (h == 0) ? 1 : 0);
  }

  k_final<<<nElemBlocks, 256, 0, stream>>>(adj, tmask, smask, scores, wvec, thr, out);
}